// LinearUpsampling3D_29661044146297
// MI455X (gfx1250) — compile-verified
//
#include <hip/hip_runtime.h>

// ---------------------------------------------------------------------------
// 2x trilinear-style transposed conv (lhs_dilation=2, taps [0.4 0.8 0.8 0.4])
// input  (2,64,48,48,48) f32  ->  output (2,64,96,96,96) f32
//
// Per axis:  out[2i]   = 0.4*in[i-1] + 0.8*in[i]
//            out[2i+1] = 0.8*in[i]   + 0.4*in[i+1]   (OOB inputs = 0)
//
// Memory-bound kernel (~510 MB traffic, ~1.8 GFLOP): one fused pass,
// input tile staged to LDS via CDNA5 async global->LDS loads, coalesced
// float4 output stores.
// ---------------------------------------------------------------------------

#define IN_D     48
#define OUT_D    96
#define TILE_D   8            // output tile extent in D
#define TILE_H   8            // output tile extent in H (W is full 96)
#define LDS_D    6            // (TILE_D/2 + 2) input halo tile
#define LDS_H    6
#define LDS_W    50           // (48/… full W inputs + 2 halo)
#define LDS_N    (LDS_D * LDS_H * LDS_W)   // 1800 floats = 7.2 KB
#define NTHREADS 256
#define QUADS_PER_ROW 24      // 96 W-outputs = 24 float4 per (d,h) row
#define WORK_ITEMS (TILE_D * TILE_H * QUADS_PER_ROW)   // 1536 float4 stores

__device__ __forceinline__ void async_g2lds_b32(unsigned lds_byte_off,
                                                const float* gaddr) {
  // CDNA5 async copy: VDST = LDS byte address VGPR, VADDR = 64-bit global
  // address, saddr disabled. Tracked by ASYNCcnt.
  asm volatile("global_load_async_to_lds_b32 %0, %1, off"
               :: "v"(lds_byte_off), "v"(gaddr)
               : "memory");
}

__device__ __forceinline__ void wait_asynccnt0() {
#if __has_builtin(__builtin_amdgcn_s_wait_asynccnt)
  __builtin_amdgcn_s_wait_asynccnt(0);
#else
  asm volatile("s_wait_asynccnt 0" ::: "memory");
#endif
}

__global__ __launch_bounds__(NTHREADS)
void LinearUpsampling3D_kernel(const float* __restrict__ in,
                               float* __restrict__ out) {
  __shared__ float tile[LDS_N];

  const int nc = blockIdx.z;                 // fused (n,c) index, 0..127
  const int d0 = blockIdx.y * TILE_D;        // output tile origin (even)
  const int h0 = blockIdx.x * TILE_H;
  const int base_di = (d0 >> 1) - 1;         // input halo tile origin
  const int base_hi = (h0 >> 1) - 1;
  const float* __restrict__ in_nc =
      in + (size_t)nc * (IN_D * IN_D * IN_D);

  // Low 32 bits of the flat shared address == LDS byte offset on gfx1250
  // (shared aperture lives in addr[63:32]).
  const unsigned lds_base = (unsigned)(unsigned long long)(void*)tile;

  // ---- Stage input halo tile: async loads for in-range, DS zero for OOB ----
  for (int e = threadIdx.x; e < LDS_N; e += NTHREADS) {
    const int lw = e % LDS_W;
    const int t  = e / LDS_W;
    const int lh = t % LDS_H;
    const int ld = t / LDS_H;
    const int gd = base_di + ld;
    const int gh = base_hi + lh;
    const int gw = lw - 1;
    if ((unsigned)gd < (unsigned)IN_D &&
        (unsigned)gh < (unsigned)IN_D &&
        (unsigned)gw < (unsigned)IN_D) {
      const float* gp = in_nc + ((size_t)gd * IN_D + gh) * IN_D + gw;
      async_g2lds_b32(lds_base + (unsigned)(e * 4), gp);
    } else {
      tile[e] = 0.0f;                         // halo outside tensor -> 0
    }
  }
  wait_asynccnt0();          // our wave's async copies landed in LDS
  __syncthreads();           // DS zero-stores + cross-wave visibility

  // ---- Compute: each thread emits 6 float4 (4 consecutive W outputs) ----
  const float2* __restrict__ T2 = (const float2*)tile;   // rows are 25 float2

  #pragma unroll
  for (int k = 0; k < WORK_ITEMS / NTHREADS; ++k) {
    const int flat = threadIdx.x + k * NTHREADS;   // < 1536
    const int q  = flat % QUADS_PER_ROW;           // W quad: outputs 4q..4q+3
    const int r  = flat / QUADS_PER_ROW;           // 0..63
    const int oh = r & (TILE_H - 1);
    const int od = r >> 3;

    // Per-axis tap selection (tile origin is even, so parity(od)==parity(do))
    int dA; float wdA, wdB;
    if (od & 1) { dA = (od >> 1) + 1; wdA = 0.8f; wdB = 0.4f; }
    else        { dA =  od >> 1;      wdA = 0.4f; wdB = 0.8f; }
    int hA; float whA, whB;
    if (oh & 1) { hA = (oh >> 1) + 1; whA = 0.8f; whB = 0.4f; }
    else        { hA =  oh >> 1;      whA = 0.4f; whB = 0.8f; }
    const int dB = dA + 1, hB = hA + 1;

    const float w00 = wdA * whA, w01 = wdA * whB;
    const float w10 = wdB * whA, w11 = wdB * whB;

    // W-input columns lw = 2q .. 2q+3  ==  float2 elements q and q+1 per row
    const int rAA = (dA * LDS_H + hA) * (LDS_W / 2) + q;
    const int rAB = (dA * LDS_H + hB) * (LDS_W / 2) + q;
    const int rBA = (dB * LDS_H + hA) * (LDS_W / 2) + q;
    const int rBB = (dB * LDS_H + hB) * (LDS_W / 2) + q;

    const float2 pAA = T2[rAA],     pAB = T2[rAB];
    const float2 pBA = T2[rBA],     pBB = T2[rBB];
    const float2 sAA = T2[rAA + 1], sAB = T2[rAB + 1];
    const float2 sBA = T2[rBA + 1], sBB = T2[rBB + 1];

    // D/H-combined W row values T(lw) for lw = 2q..2q+3
    const float t0 = w00 * pAA.x + w01 * pAB.x + w10 * pBA.x + w11 * pBB.x;
    const float t1 = w00 * pAA.y + w01 * pAB.y + w10 * pBA.y + w11 * pBB.y;
    const float t2 = w00 * sAA.x + w01 * sAB.x + w10 * sBA.x + w11 * sBB.x;
    const float t3 = w00 * sAA.y + w01 * sAB.y + w10 * sBA.y + w11 * sBB.y;

    // W-axis 2-tap blend: outputs 4q..4q+3 use input cols (i-1,i,i+1,i+2)
    float4 o;
    o.x = 0.4f * t0 + 0.8f * t1;
    o.y = 0.8f * t1 + 0.4f * t2;
    o.z = 0.4f * t1 + 0.8f * t2;
    o.w = 0.8f * t2 + 0.4f * t3;

    const size_t oidx =
        (((size_t)nc * OUT_D + (d0 + od)) * OUT_D + (h0 + oh)) * OUT_D + 4 * q;
    *(float4*)(out + oidx) = o;                 // 16B-aligned coalesced store
  }
}

extern "C" void kernel_launch(void* const* d_in, const int* in_sizes, int n_in,
                              void* d_out, int out_size, void* d_ws, size_t ws_size,
                              hipStream_t stream) {
  const float* inp = (const float*)d_in[0];
  float* out = (float*)d_out;

  const int nc_total = in_sizes[0] / (IN_D * IN_D * IN_D);   // 2*64 = 128
  dim3 grid(OUT_D / TILE_H, OUT_D / TILE_D, nc_total);       // (12,12,128)
  dim3 block(NTHREADS);

  LinearUpsampling3D_kernel<<<grid, block, 0, stream>>>(inp, out);
}